// Decodering_model_4655744549220
// MI455X (gfx1250) — compile-verified
//
#include <hip/hip_runtime.h>
#include <math.h>

typedef __attribute__((ext_vector_type(2))) float v2f;
typedef __attribute__((ext_vector_type(8))) float v8f;

#define BATCH 2048
#define NCOL 127
#define MROW 63
#define NITER 8
#define BIGV 1e30f
#define OUT_SLICE (BATCH * NCOL)
#define CV_STRIDE 129   // 64 rows x 129 floats: bank-conflict-free stores & WMMA feeds

// One block per codeword. 64 threads = 2 wave32s.
__global__ __launch_bounds__(64) void bp_decode_kernel(
    const float* __restrict__ soft_in,   // [B,127]
    const int*   __restrict__ labels,    // [B,127]
    const int*   __restrict__ H,         // [63,127]
    const float* __restrict__ cw,        // [1]
    float* __restrict__ out,             // [9,B,127]
    float* __restrict__ part_ce,         // [B]
    float* __restrict__ part_mse)        // [B]
{
    __shared__ float sh_cv[64 * CV_STRIDE];   // reconstructed cv tile (rows 63.. padded zero)
    __shared__ float sh_total[128];
    __shared__ float sh_soft[128];
    __shared__ float sh_red[128];

    const int tid  = threadIdx.x;
    const int b    = blockIdx.x;
    const int lane = tid & 31;
    const int wave = tid >> 5;
    const int m    = tid;                 // check-row owned by this thread (if < 63)

    // softplus(check_weight)
    float wv = cw[0];
    float sp = (wv > 20.f) ? wv : log1pf(expf(wv));

    // ---- Stage H through LDS (coalesced), build per-row 127-bit support masks ----
    int* shH = (int*)sh_cv;
    for (int i = tid; i < MROW * NCOL; i += 64) shH[i] = H[i];
    __syncthreads();

    unsigned hwv[4] = {0u, 0u, 0u, 0u};
    if (m < MROW) {
#pragma unroll
        for (int w_ = 0; w_ < 4; ++w_) {
            const int lim = (w_ == 3) ? 31 : 32;
            unsigned acc_ = 0u;
            for (int b_ = 0; b_ < lim; ++b_)
                acc_ |= (((unsigned)shH[m * NCOL + (w_ << 5) + b_]) & 1u) << b_;
            hwv[w_] = acc_;
        }
    }
    __syncthreads();

    // ---- Zero the cv tile (padding rows/cols stay zero forever) ----
    for (int i = tid; i < 64 * CV_STRIDE; i += 64) sh_cv[i] = 0.f;

    // ---- Load soft input; total_0 = soft_input; write outs[0] ----
    for (int n = tid; n < NCOL; n += 64) {
        float v = soft_in[b * NCOL + n];
        sh_soft[n]  = v;
        sh_total[n] = v;
        out[b * NCOL + n] = v;            // outs[0]
    }
    if (tid == 0) { sh_soft[127] = 0.f; sh_total[127] = 0.f; }
    __syncthreads();

    // ---- Register-resident per-row message state ----
    float oa1 = 0.f, oa2 = 0.f;           // sp * min1/min2 * rowsign (signed)
    int   oidx = -1;                      // column holding the row minimum
    unsigned onv[4] = {0u,0u,0u,0u};      // per-entry sign(vc) < 0 bits
    unsigned ozv[4] = {0u,0u,0u,0u};      // per-entry sign(vc) == 0 bits

    for (int it = 0; it < NITER; ++it) {
        // ================= Phase B: per-check-row min-sum update =================
        if (m < MROW) {
            float mn1 = 3.0e38f, mn2 = 3.0e38f;
            int   nidx = -1;
            unsigned nnv[4], nzv[4];
            unsigned pneg = 0u, pzer = 0u;
#pragma unroll
            for (int w_ = 0; w_ < 4; ++w_) {
                const int lim = (w_ == 3) ? 31 : 32;
                const unsigned hb = hwv[w_], ob = onv[w_], oz = ozv[w_];
                unsigned nn = 0u, nz = 0u;
                for (int b_ = 0; b_ < lim; ++b_) {
                    if ((hb >> b_) & 1u) {
                        const int n = (w_ << 5) | b_;
                        const float tot = sh_total[n];
                        // reconstruct old cv[m,n] from compact state
                        float mag = (n == oidx) ? oa2 : oa1;
                        float cvo = ((oz >> b_) & 1u) ? 0.f
                                   : (((ob >> b_) & 1u) ? -mag : mag);
                        float vc = tot - cvo;
                        float a  = fminf(fabsf(vc), BIGV);
                        unsigned ng = (vc < 0.f)  ? 1u : 0u;
                        unsigned zr = (vc == 0.f) ? 1u : 0u;
                        nn |= ng << b_;  nz |= zr << b_;
                        pneg ^= ng;      pzer |= zr;
                        if (a < mn1)      { mn2 = mn1; mn1 = a; nidx = n; }
                        else if (a < mn2) { mn2 = a; }
                    }
                }
                nnv[w_] = nn; nzv[w_] = nz;
            }
            const float rs  = pzer ? 0.f : (pneg ? -1.f : 1.f);
            const float na1 = sp * mn1 * rs;
            const float na2 = sp * mn2 * rs;
            // materialize new cv row into the LDS tile (support entries only)
#pragma unroll
            for (int w_ = 0; w_ < 4; ++w_) {
                const int lim = (w_ == 3) ? 31 : 32;
                const unsigned hb = hwv[w_], nn = nnv[w_], nz = nzv[w_];
                for (int b_ = 0; b_ < lim; ++b_) {
                    if ((hb >> b_) & 1u) {
                        const int n = (w_ << 5) | b_;
                        float mag = (n == nidx) ? na2 : na1;
                        float v = ((nz >> b_) & 1u) ? 0.f
                                 : (((nn >> b_) & 1u) ? -mag : mag);
                        sh_cv[m * CV_STRIDE + n] = v;
                    }
                }
            }
            oa1 = na1; oa2 = na2; oidx = nidx;
#pragma unroll
            for (int w_ = 0; w_ < 4; ++w_) { onv[w_] = nnv[w_]; ozv[w_] = nzv[w_]; }
        }
        __syncthreads();

        // ====== Phase A: WMMA marginalization total[n] = soft[n] + sum_m cv[m,n] ======
        // 16 rank-4 f32 WMMA updates with a ones A-matrix sum 64 rows exactly.
        // All lanes of both waves execute (EXEC must be all-ones for WMMA).
        {
            const int half = lane >> 4;
            for (int t = 0; t < 4; ++t) {
                const int tile = wave * 4 + t;           // 8 column tiles of 16
                const int col  = tile * 16 + (lane & 15);
                v8f acc = {0.f,0.f,0.f,0.f,0.f,0.f,0.f,0.f};
                v2f ones; ones[0] = 1.f; ones[1] = 1.f;
#pragma unroll
                for (int j = 0; j < 16; ++j) {
                    v2f bv;
                    bv[0] = sh_cv[(4 * j + half)     * CV_STRIDE + col];
                    bv[1] = sh_cv[(4 * j + 2 + half) * CV_STRIDE + col];
                    acc = __builtin_amdgcn_wmma_f32_16x16x4_f32(
                        false, ones, false, bv, (short)0, acc, false, false);
                }
                // every row of D equals the column sum; element 0 suffices
                const float total = sh_soft[col] + acc[0];
                if (lane < 16) {
                    sh_total[col] = total;
                    if (col < NCOL)
                        out[(size_t)(it + 1) * OUT_SLICE + b * NCOL + col] = total;
                }
            }
        }
        __syncthreads();

        // ====== Loss partials from outs[2] (produced at it==1) ======
        if (it == 1) {
            float ce = 0.f, mse = 0.f;
            for (int n = tid; n < NCOL; n += 64) {
                const float z  = -sh_total[n];
                const float lf = (float)labels[b * NCOL + n];
                ce += fmaxf(z, 0.f) - z * lf + log1pf(expf(-fabsf(z)));
                const float p = 1.f / (1.f + expf(-z));
                const float d = p - lf;
                mse += d * d;
            }
            sh_red[tid] = ce; sh_red[64 + tid] = mse;
            __syncthreads();
            if (tid == 0) {
                float sc = 0.f, sm = 0.f;
                for (int i = 0; i < 64; ++i) { sc += sh_red[i]; sm += sh_red[64 + i]; }
                part_ce[b]  = sc;             // overwritten each call -> deterministic
                part_mse[b] = sm;
            }
            __syncthreads();
        }
    }
}

// Deterministic fixed-order cross-block reduction -> scalar loss.
__global__ __launch_bounds__(256) void loss_finalize_kernel(
    const float* __restrict__ part_ce,
    const float* __restrict__ part_mse,
    float* __restrict__ out_loss)
{
    __shared__ float rc[256], rm[256];
    const int t = threadIdx.x;
    float sc = 0.f, sm = 0.f;
    for (int i = t; i < BATCH; i += 256) { sc += part_ce[i]; sm += part_mse[i]; }
    rc[t] = sc; rm[t] = sm;
    __syncthreads();
    for (int s = 128; s > 0; s >>= 1) {
        if (t < s) { rc[t] += rc[t + s]; rm[t] += rm[t + s]; }
        __syncthreads();
    }
    if (t == 0) {
        const float ce_mean = rc[0] / (float)(BATCH * NCOL);
        out_loss[0] = 0.5f * ce_mean + 0.5f * 0.01f * rm[0];
    }
}

extern "C" void kernel_launch(void* const* d_in, const int* in_sizes, int n_in,
                              void* d_out, int out_size, void* d_ws, size_t ws_size,
                              hipStream_t stream) {
    (void)in_sizes; (void)n_in; (void)out_size; (void)ws_size;
    const float* soft   = (const float*)d_in[0];
    const int*   labels = (const int*)d_in[1];
    const int*   H      = (const int*)d_in[2];
    const float* cw     = (const float*)d_in[3];
    float* out      = (float*)d_out;
    float* part_ce  = (float*)d_ws;          // [2048]
    float* part_mse = part_ce + BATCH;       // [2048]

    bp_decode_kernel<<<BATCH, 64, 0, stream>>>(soft, labels, H, cw, out, part_ce, part_mse);
    loss_finalize_kernel<<<1, 256, 0, stream>>>(part_ce, part_mse, out + (size_t)9 * OUT_SLICE);
}